// ThreeDInteraction_39891656245705
// MI455X (gfx1250) — compile-verified
//
#include <hip/hip_runtime.h>
#include <hip/hip_bf16.h>

typedef __attribute__((ext_vector_type(2))) float v2f;
typedef __attribute__((ext_vector_type(4))) float v4f;
typedef __attribute__((ext_vector_type(8))) float v8f;

#define DIMF 64

// ---------------------------------------------------------------------------
// Utility: zero an int array (needed every call; harness doesn't re-poison)
// ---------------------------------------------------------------------------
__global__ __launch_bounds__(256)
void zero_i32_kernel(int* __restrict__ p, int n) {
  int i = blockIdx.x * blockDim.x + threadIdx.x;
  if (i < n) p[i] = 0;
}

// ---------------------------------------------------------------------------
// updated = sigmoid(atom @ W_update + b_update)   (n_rows x 64) @ (64 x 64)
// One wave (32 lanes) computes a 16x64 output stripe with fp32 WMMA.
// A layout (16x4 f32): lane = 16*(k/2) + M ; v0 = K=k0, v1 = K=k0+1
// B layout (4x16 f32): mirrored; C/D: VGPR r -> M = r + 8*(lane/16), N = lane%16
// ---------------------------------------------------------------------------
__global__ __launch_bounds__(256)
void atom_update_kernel(const float* __restrict__ atom, const float* __restrict__ W,
                        const float* __restrict__ bias, float* __restrict__ updated,
                        int n_rows) {
  const int lane = threadIdx.x & 31;
  const int wave = threadIdx.x >> 5;
  const int tile = blockIdx.x * (blockDim.x >> 5) + wave;
  const int row0 = tile << 4;
  if (row0 >= n_rows) return;

  const int mrow = lane & 15;          // M for A-loads, N for B/C
  const int kq   = (lane >> 4) << 1;   // 0 or 2 (K sub-offset per lane half)
  int arow = row0 + mrow;
  if (arow >= n_rows) arow = n_rows - 1;   // clamp: keeps EXEC all-1 for WMMA
  const float* arow_ptr = atom + (size_t)arow * DIMF;

  v8f acc[4] = {};
  for (int kk = 0; kk < DIMF; kk += 4) {
    v2f a = *(const v2f*)(arow_ptr + kk + kq);
#pragma unroll
    for (int nt = 0; nt < 4; ++nt) {
      v2f b;
      b.x = W[(size_t)(kk + kq)     * DIMF + nt * 16 + mrow];
      b.y = W[(size_t)(kk + kq + 1) * DIMF + nt * 16 + mrow];
      acc[nt] = __builtin_amdgcn_wmma_f32_16x16x4_f32(
          false, a, false, b, (short)0, acc[nt], false, false);
    }
  }

  const int mhi = (lane >> 4) << 3;    // +8 rows for upper lane half
#pragma unroll
  for (int nt = 0; nt < 4; ++nt) {
    const float bv = bias[nt * 16 + mrow];
#pragma unroll
    for (int r = 0; r < 8; ++r) {
      const int row = row0 + r + mhi;
      if (row < n_rows) {
        float x = acc[nt][r] + bv;
        // sigmoid via hw transcendentals: v_exp_f32 + v_rcp_f32
        // (avoids the ~10-op IEEE v_div_scale/fixup sequence)
        x = __builtin_amdgcn_rcpf(1.0f + __expf(-x));
        updated[(size_t)row * DIMF + nt * 16 + mrow] = x;
      }
    }
  }
}

// ---------------------------------------------------------------------------
// Histogram of triples per destination bond (segment)
// ---------------------------------------------------------------------------
__global__ __launch_bounds__(256)
void hist_kernel(const int* __restrict__ tbi, int* __restrict__ counts, int n_triples) {
  int t = blockIdx.x * blockDim.x + threadIdx.x;
  if (t < n_triples) atomicAdd(&counts[tbi[2 * t]], 1);
}

// ---------------------------------------------------------------------------
// Hierarchical exclusive scan over counts (chunk = 1024 per block)
// ---------------------------------------------------------------------------
__global__ __launch_bounds__(256)
void scan_reduce_kernel(const int* __restrict__ counts, int* __restrict__ bsum, int n) {
  __shared__ int sh[256];
  const int base = blockIdx.x * 1024 + threadIdx.x * 4;
  int s = 0;
#pragma unroll
  for (int i = 0; i < 4; ++i) {
    int idx = base + i;
    if (idx < n) s += counts[idx];
  }
  sh[threadIdx.x] = s;
  __syncthreads();
  for (int off = 128; off > 0; off >>= 1) {
    if (threadIdx.x < off) sh[threadIdx.x] += sh[threadIdx.x + off];
    __syncthreads();
  }
  if (threadIdx.x == 0) bsum[blockIdx.x] = sh[0];
}

__global__ __launch_bounds__(1024)
void scan_top_kernel(int* __restrict__ bsum, int nb) {   // nb <= 1024
  __shared__ int sh[1024];
  const int t = threadIdx.x;
  const int v = (t < nb) ? bsum[t] : 0;
  sh[t] = v;
  __syncthreads();
  for (int off = 1; off < 1024; off <<= 1) {
    int add = (t >= off) ? sh[t - off] : 0;
    __syncthreads();
    sh[t] += add;
    __syncthreads();
  }
  if (t < nb) bsum[t] = sh[t] - v;   // exclusive
}

__global__ __launch_bounds__(256)
void scan_final_kernel(const int* __restrict__ counts, const int* __restrict__ boff,
                       int* __restrict__ offsets, int* __restrict__ cursor, int n) {
  __shared__ int sh[256];
  const int tid = threadIdx.x;
  const int base = blockIdx.x * 1024 + tid * 4;
  int v0 = 0, v1 = 0, v2 = 0, v3 = 0;
  if (base + 0 < n) v0 = counts[base + 0];
  if (base + 1 < n) v1 = counts[base + 1];
  if (base + 2 < n) v2 = counts[base + 2];
  if (base + 3 < n) v3 = counts[base + 3];
  const int s = v0 + v1 + v2 + v3;
  sh[tid] = s;
  __syncthreads();
  for (int off = 1; off < 256; off <<= 1) {
    int add = (tid >= off) ? sh[tid - off] : 0;
    __syncthreads();
    sh[tid] += add;
    __syncthreads();
  }
  int excl = sh[tid] - s + boff[blockIdx.x];
  const int e0 = excl, e1 = e0 + v0, e2 = e1 + v1, e3 = e2 + v2;
  if (base + 0 < n) { offsets[base + 0] = e0; cursor[base + 0] = e0; }
  if (base + 1 < n) { offsets[base + 1] = e1; cursor[base + 1] = e1; }
  if (base + 2 < n) { offsets[base + 2] = e2; cursor[base + 2] = e2; }
  if (base + 3 < n) { offsets[base + 3] = e3; cursor[base + 3] = e3; }
}

// ---------------------------------------------------------------------------
// Scatter triple IDs into CSR list (cursor pre-initialized to offsets)
// ---------------------------------------------------------------------------
__global__ __launch_bounds__(256)
void fill_list_kernel(const int* __restrict__ tbi, int* __restrict__ cursor,
                      int* __restrict__ list, int n_triples) {
  int t = blockIdx.x * blockDim.x + threadIdx.x;
  if (t >= n_triples) return;
  int pos = atomicAdd(&cursor[tbi[2 * t]], 1);
  list[pos] = t;
}

// ---------------------------------------------------------------------------
// Per-bond gather-accumulate: summed[b] = sum_{t in CSR(b)} basis[t]*updated[third(t)]
// 16 lanes per bond, 4 floats (one b128) per lane. Zero atomics; register acc.
// basis is streamed non-temporally (512 MB, touched once) to keep L2 for
// the 12.8 MB `updated` table and the 128 MB output.
// ---------------------------------------------------------------------------
__global__ __launch_bounds__(256)
void gather_accumulate_kernel(const float* __restrict__ basis, const float* __restrict__ updated,
                              const int* __restrict__ tbi, const int* __restrict__ bai,
                              const int* __restrict__ offsets, const int* __restrict__ counts,
                              const int* __restrict__ list, float* __restrict__ summed,
                              int n_bonds) {
  const int gid = blockIdx.x * blockDim.x + threadIdx.x;
  const int b = gid >> 4;
  if (b >= n_bonds) return;
  const int q = (gid & 15) << 2;
  const int start = offsets[b];
  const int cnt = counts[b];
  v4f acc = {0.f, 0.f, 0.f, 0.f};
  for (int j = 0; j < cnt; ++j) {
    const int t = list[start + j];
    const int b1 = tbi[2 * t + 1];
    const int third = bai[2 * b1 + 1];
    v4f bs = __builtin_nontemporal_load((const v4f*)(basis + (size_t)t * DIMF + q));
    v4f up = *(const v4f*)(updated + (size_t)third * DIMF + q);
    acc += bs * up;
  }
  *(v4f*)(summed + (size_t)b * DIMF + q) = acc;   // plain store, no zero-init needed
}

// ---------------------------------------------------------------------------
// out = bond + summed @ W_fuse + b_fuse  — in place on d_out (holds summed).
// Row m of the result depends only on row m of summed; each wave reads its
// 16 rows fully before storing them, so in-place is safe.
// ---------------------------------------------------------------------------
__global__ __launch_bounds__(256)
void fuse_kernel(float* __restrict__ out, const float* __restrict__ bond,
                 const float* __restrict__ W, const float* __restrict__ bias,
                 int n_rows) {
  const int lane = threadIdx.x & 31;
  const int wave = threadIdx.x >> 5;
  const int tile = blockIdx.x * (blockDim.x >> 5) + wave;
  const int row0 = tile << 4;
  if (row0 >= n_rows) return;

  const int mrow = lane & 15;
  const int kq   = (lane >> 4) << 1;
  int arow = row0 + mrow;
  if (arow >= n_rows) arow = n_rows - 1;
  const float* arow_ptr = out + (size_t)arow * DIMF;

  v8f acc[4] = {};
  for (int kk = 0; kk < DIMF; kk += 4) {
    v2f a = *(const v2f*)(arow_ptr + kk + kq);
#pragma unroll
    for (int nt = 0; nt < 4; ++nt) {
      v2f b;
      b.x = W[(size_t)(kk + kq)     * DIMF + nt * 16 + mrow];
      b.y = W[(size_t)(kk + kq + 1) * DIMF + nt * 16 + mrow];
      acc[nt] = __builtin_amdgcn_wmma_f32_16x16x4_f32(
          false, a, false, b, (short)0, acc[nt], false, false);
    }
  }

  const int mhi = (lane >> 4) << 3;
#pragma unroll
  for (int nt = 0; nt < 4; ++nt) {
    const float bv = bias[nt * 16 + mrow];
#pragma unroll
    for (int r = 0; r < 8; ++r) {
      const int row = row0 + r + mhi;
      if (row < n_rows) {
        const size_t idx = (size_t)row * DIMF + nt * 16 + mrow;
        out[idx] = acc[nt][r] + bv + bond[idx];
      }
    }
  }
}

// ---------------------------------------------------------------------------
extern "C" void kernel_launch(void* const* d_in, const int* in_sizes, int n_in,
                              void* d_out, int out_size, void* d_ws, size_t ws_size,
                              hipStream_t stream) {
  const float* atom     = (const float*)d_in[0];
  const float* bond     = (const float*)d_in[1];
  const float* basis    = (const float*)d_in[2];
  const int*   bai      = (const int*)d_in[3];   // bond_atom_indices (n_bonds,2)
  const int*   tbi      = (const int*)d_in[4];   // triple_bond_indices (n_triples,2)
  const float* W_update = (const float*)d_in[5];
  const float* b_update = (const float*)d_in[6];
  const float* W_fuse   = (const float*)d_in[7];
  const float* b_fuse   = (const float*)d_in[8];
  float* out = (float*)d_out;

  const int n_atoms   = in_sizes[0] / DIMF;
  const int n_bonds   = in_sizes[1] / DIMF;
  const int n_triples = in_sizes[2] / DIMF;

  // Workspace carve-out (~27 MB): updated table + CSR arrays
  char* ws = (char*)d_ws;
  size_t off = 0;
  auto alloc = [&](size_t bytes) -> void* {
    off = (off + 255) & ~(size_t)255;
    void* p = ws + off;
    off += bytes;
    return p;
  };
  float* updated = (float*)alloc((size_t)n_atoms * DIMF * sizeof(float));
  int* counts  = (int*)alloc((size_t)n_bonds * sizeof(int));
  int* offsets = (int*)alloc((size_t)n_bonds * sizeof(int));
  int* cursor  = (int*)alloc((size_t)n_bonds * sizeof(int));
  int* bsum    = (int*)alloc(1024 * sizeof(int));
  int* list    = (int*)alloc((size_t)n_triples * sizeof(int));

  // 1. zero histogram (must happen every call: graph replay determinism)
  zero_i32_kernel<<<(n_bonds + 255) / 256, 256, 0, stream>>>(counts, n_bonds);

  // 2. updated = sigmoid(atom @ W_update + b) — 50k rows instead of 2M (40x less GEMM)
  {
    const int tiles = (n_atoms + 15) / 16;
    atom_update_kernel<<<(tiles + 7) / 8, 256, 0, stream>>>(atom, W_update, b_update,
                                                            updated, n_atoms);
  }

  // 3-7. Build CSR triple lists per bond (replaces 128M fp32 atomics with streaming)
  hist_kernel<<<(n_triples + 255) / 256, 256, 0, stream>>>(tbi, counts, n_triples);
  const int nb = (n_bonds + 1023) / 1024;   // <= 1024 for n_bonds <= 1M
  scan_reduce_kernel<<<nb, 256, 0, stream>>>(counts, bsum, n_bonds);
  scan_top_kernel<<<1, 1024, 0, stream>>>(bsum, nb);
  scan_final_kernel<<<nb, 256, 0, stream>>>(counts, bsum, offsets, cursor, n_bonds);
  fill_list_kernel<<<(n_triples + 255) / 256, 256, 0, stream>>>(tbi, cursor, list, n_triples);

  // 8. summed (in d_out) = segment-gather of basis * updated[third]
  {
    const long threads = (long)n_bonds * 16;
    gather_accumulate_kernel<<<(threads + 255) / 256, 256, 0, stream>>>(
        basis, updated, tbi, bai, offsets, counts, list, out, n_bonds);
  }

  // 9. out = bond + summed @ W_fuse + b_fuse (in place, WMMA)
  {
    const int tiles = (n_bonds + 15) / 16;
    fuse_kernel<<<(tiles + 7) / 8, 256, 0, stream>>>(out, bond, W_fuse, b_fuse, n_bonds);
  }
}